// FCx2DetHead_27006754357668
// MI455X (gfx1250) — compile-verified
//
#include <hip/hip_runtime.h>

// ---------------------------------------------------------------------------
// Deformable ROI pool + FC heads for MI455X (gfx1250), fp32 WMMA path.
// GEMMs on V_WMMA_F32_16X16X4_F32, wave tile 32x64, software-pipelined K loop
// enforced with __builtin_amdgcn_sched_barrier so next-iteration fragment
// loads stay in flight while current wmmas execute (no s_wait_loadcnt 0x0
// serialization). Ragged weights zero-padded so the K loop is branch-free.
// ---------------------------------------------------------------------------

#define CCH   256
#define HH    25
#define WW    25
#define BBATCH 8
#define NROI  256
#define OUTD  7
#define NBIN  49            // 7*7
#define KIN   12544         // 256*49
#define SSC   (25.0f / 255.0f)
#define TRANS_STD 0.1f

typedef float v2f __attribute__((ext_vector_type(2)));
typedef float v8f __attribute__((ext_vector_type(8)));

#define WMMA_F32(A_, B_, C_) \
    __builtin_amdgcn_wmma_f32_16x16x4_f32(false, (A_), false, (B_), (short)0, (C_), false, false)

// ---- NCHW -> NHWC transpose so channel gathers are coalesced --------------
__global__ void nchw_to_nhwc_k(const float* __restrict__ x, float* __restrict__ feat) {
    int c   = threadIdx.x;                 // 0..255
    int bhw = blockIdx.x;                  // 0..(8*625-1)
    int b   = bhw / (HH * WW);
    int hw  = bhw - b * (HH * WW);
    feat[(size_t)(b * (HH * WW) + hw) * CCH + c] =
        x[((size_t)(b * CCH + c)) * (HH * WW) + hw];
}

// ---- zero-pad a row-major (Nsrc x K) matrix into (Npad x K) ---------------
__global__ void pad_rows_k(const float* __restrict__ src, float* __restrict__ dst,
                           int Nsrc, int K) {
    size_t i = (size_t)blockIdx.x * blockDim.x + threadIdx.x;  // over Npad*K
    int n = (int)(i / (size_t)K);
    int k = (int)(i - (size_t)n * K);
    dst[i] = (n < Nsrc) ? src[(size_t)n * K + k] : 0.0f;
}

// ---- Deformable ROI align-pool; block = (roi, bin), thread = channel ------
__global__ void droi_pool_k(const float* __restrict__ feat,
                            const float* __restrict__ rois,
                            const float* __restrict__ off, int use_off,
                            float* __restrict__ out) {
    int nb  = blockIdx.x;
    int n   = nb / NBIN;
    int bin = nb - n * NBIN;
    int ph  = bin / OUTD;
    int pw  = bin - ph * OUTD;
    int c   = threadIdx.x;

    const float* r = rois + n * 5;
    int   b  = (int)r[0];
    float sw = rintf(r[1]) * SSC - 0.5f;                 // rintf == round-half-even
    float sh = rintf(r[2]) * SSC - 0.5f;
    float rw = fmaxf((rintf(r[3]) + 1.0f) * SSC - 0.5f - sw, 0.1f);
    float rh = fmaxf((rintf(r[4]) + 1.0f) * SSC - 0.5f - sh, 0.1f);
    float bw = rw * (1.0f / OUTD);
    float bh = rh * (1.0f / OUTD);

    float tx = 0.0f, ty = 0.0f;
    if (use_off) {                                       // off: (N, 2, 7, 7)
        tx = off[n * (2 * NBIN) + bin]        * TRANS_STD;
        ty = off[n * (2 * NBIN) + NBIN + bin] * TRANS_STD;
    }
    float wstart = (float)pw * bw + sw + tx * rw;
    float hstart = (float)ph * bh + sh + ty * rh;
    float swx = bw * 0.25f, swy = bh * 0.25f;

    const float* fb = feat + (size_t)b * (HH * WW * CCH);
    float acc = 0.0f;
    int   cnt = 0;
    #pragma unroll
    for (int sy = 0; sy < 4; ++sy) {
        float syf = hstart + (float)sy * swy;
        #pragma unroll
        for (int sx = 0; sx < 4; ++sx) {
            float sxf = wstart + (float)sx * swx;
            // validity is channel-uniform -> no divergence within the block
            bool valid = (sxf >= -0.5f) && (sxf <= (float)WW - 0.5f) &&
                         (syf >= -0.5f) && (syf <= (float)HH - 0.5f);
            if (!valid) continue;
            ++cnt;
            float xc = fminf(fmaxf(sxf, 0.0f), (float)(WW - 1));
            float yc = fminf(fmaxf(syf, 0.0f), (float)(HH - 1));
            float x0 = floorf(xc), y0 = floorf(yc);
            float dx = xc - x0,   dy = yc - y0;
            int xi0 = (int)x0, yi0 = (int)y0;
            int xi1 = (int)ceilf(xc), yi1 = (int)ceilf(yc);
            float v00 = fb[((size_t)(yi0 * WW + xi0)) * CCH + c];
            float v01 = fb[((size_t)(yi0 * WW + xi1)) * CCH + c];
            float v10 = fb[((size_t)(yi1 * WW + xi0)) * CCH + c];
            float v11 = fb[((size_t)(yi1 * WW + xi1)) * CCH + c];
            acc += (1.0f - dx) * (1.0f - dy) * v00 + dx * (1.0f - dy) * v01 +
                   (1.0f - dx) * dy * v10 + dx * dy * v11;
        }
    }
    out[(size_t)n * KIN + c * NBIN + bin] = acc / fmaxf((float)cnt, 1.0f);
}

// ---- fp32 WMMA GEMM:  out[M=256, N] = A[M,K] * Wp[Npad,K]^T + bias (+ReLU)
// One wave per block; wave tile 32(M) x 64(N) -> eight v8f accumulators.
// Wp must have >= gridDim.x*64 zero-padded rows (K loop needs no guards).
// Software pipeline: load k+4 fragments, sched_barrier, then the 8 wmmas for
// k, so loads stay outstanding across the matrix ops.
// Fragment lane layout per ISA 7.12.2:
//   A 16x4 : lanes 0-15 hold M=lane,K={k,k+1}; lanes 16-31 hold K={k+2,k+3}
//   B 4x16 : lanes 0-15 hold N=lane,K={k,k+1}; lanes 16-31 hold K={k+2,k+3}
__global__ void __launch_bounds__(32)
gemm_wmma_f32_k(const float* __restrict__ A, const float* __restrict__ Wp,
                const float* __restrict__ bias, float* __restrict__ out,
                int N, int K, int doRelu) {
    const int lane = threadIdx.x & 31;
    const int l16  = lane & 15;
    const int kh   = (lane >> 4) << 1;          // 0 or 2

    const int mTile = blockIdx.y * 32;
    const int nBase = blockIdx.x * 64;

    const float* aP0 = A + (size_t)(mTile +  0 + l16) * K + kh;
    const float* aP1 = A + (size_t)(mTile + 16 + l16) * K + kh;
    const float* bP0 = Wp + (size_t)(nBase +  0 + l16) * K + kh;
    const float* bP1 = Wp + (size_t)(nBase + 16 + l16) * K + kh;
    const float* bP2 = Wp + (size_t)(nBase + 32 + l16) * K + kh;
    const float* bP3 = Wp + (size_t)(nBase + 48 + l16) * K + kh;

    v8f c00 = {}, c01 = {}, c02 = {}, c03 = {};
    v8f c10 = {}, c11 = {}, c12 = {}, c13 = {};

    // prologue: fragments for k = 0
    v2f a0 = *(const v2f*)(aP0);
    v2f a1 = *(const v2f*)(aP1);
    v2f b0 = *(const v2f*)(bP0);
    v2f b1 = *(const v2f*)(bP1);
    v2f b2 = *(const v2f*)(bP2);
    v2f b3 = *(const v2f*)(bP3);

    for (int k = 4; k < K; k += 4) {
        // next-iteration loads issue first and stay outstanding...
        v2f a0n = *(const v2f*)(aP0 + k);
        v2f a1n = *(const v2f*)(aP1 + k);
        v2f b0n = *(const v2f*)(bP0 + k);
        v2f b1n = *(const v2f*)(bP1 + k);
        v2f b2n = *(const v2f*)(bP2 + k);
        v2f b3n = *(const v2f*)(bP3 + k);

        __builtin_amdgcn_sched_barrier(0);   // ...while these wmmas execute

        c00 = WMMA_F32(a0, b0, c00);
        c01 = WMMA_F32(a0, b1, c01);
        c02 = WMMA_F32(a0, b2, c02);
        c03 = WMMA_F32(a0, b3, c03);
        c10 = WMMA_F32(a1, b0, c10);
        c11 = WMMA_F32(a1, b1, c11);
        c12 = WMMA_F32(a1, b2, c12);
        c13 = WMMA_F32(a1, b3, c13);

        a0 = a0n; a1 = a1n; b0 = b0n; b1 = b1n; b2 = b2n; b3 = b3n;
    }
    c00 = WMMA_F32(a0, b0, c00);
    c01 = WMMA_F32(a0, b1, c01);
    c02 = WMMA_F32(a0, b2, c02);
    c03 = WMMA_F32(a0, b3, c03);
    c10 = WMMA_F32(a1, b0, c10);
    c11 = WMMA_F32(a1, b1, c11);
    c12 = WMMA_F32(a1, b2, c12);
    c13 = WMMA_F32(a1, b3, c13);

    // C/D layout: VGPR r -> M = r (lanes 0-15) or 8+r (lanes 16-31); N = lane&15
    v8f accs[2][4] = { {c00, c01, c02, c03}, {c10, c11, c12, c13} };
    #pragma unroll
    for (int i = 0; i < 2; ++i) {
        const int mBase = mTile + i * 16 + (lane >> 4) * 8;
        #pragma unroll
        for (int t = 0; t < 4; ++t) {
            int n = nBase + t * 16 + l16;
            if (n >= N) continue;                // epilogue-only guard
            float bv = bias[n];
            #pragma unroll
            for (int r = 0; r < 8; ++r) {
                float v = accs[i][t][r] + bv;
                if (doRelu) v = fmaxf(v, 0.0f);
                out[(size_t)(mBase + r) * N + n] = v;
            }
        }
    }
}

extern "C" void kernel_launch(void* const* d_in, const int* in_sizes, int n_in,
                              void* d_out, int out_size, void* d_ws, size_t ws_size,
                              hipStream_t stream) {
    const float* x   = (const float*)d_in[0];
    const float* ro  = (const float*)d_in[1];
    const float* ow1 = (const float*)d_in[2];
    const float* ob1 = (const float*)d_in[3];
    const float* ow2 = (const float*)d_in[4];
    const float* ob2 = (const float*)d_in[5];
    const float* ow3 = (const float*)d_in[6];
    const float* ob3 = (const float*)d_in[7];
    const float* fw1 = (const float*)d_in[8];
    const float* fb1 = (const float*)d_in[9];
    const float* fw2 = (const float*)d_in[10];
    const float* fb2 = (const float*)d_in[11];
    const float* bxw = (const float*)d_in[12];
    const float* bxb = (const float*)d_in[13];
    float* out = (float*)d_out;

    // workspace layout (floats); everything is fully overwritten before read
    float* ws   = (float*)d_ws;
    float* feat = ws;                       // 8*25*25*256      = 1,280,000
    float* p    = feat + 1280000;           // 256*12544        = 3,211,264
    float* h1   = p    + 3211264;           // 256*1024         =   262,144
    float* h2   = h1   + 262144;            // 256*1024         =   262,144
    float* offb = h2   + 262144;            // 256*98           =    25,088
    float* f1   = offb + 25088;             // 256*512          =   131,072
    float* f2   = f1   + 131072;            // 256*512          =   131,072
    float* wp3  = f2   + 131072;            // 128*1024 padded ow3
    float* wpb  = wp3  + 131072;            // 64*512   padded bxw

    nchw_to_nhwc_k<<<BBATCH * HH * WW, CCH, 0, stream>>>(x, feat);

    // zero-pad ragged weights once (98x1024 -> 128x1024, 4x512 -> 64x512)
    pad_rows_k<<<(128 * 1024) / 256, 256, 0, stream>>>(ow3, wp3, 98, 1024);
    pad_rows_k<<<(64 * 512) / 256, 256, 0, stream>>>(bxw, wpb, 4, 512);

    // pass 1: zero-offset pool -> p
    droi_pool_k<<<NROI * NBIN, CCH, 0, stream>>>(feat, ro, nullptr, 0, p);

    dim3 blk(32);
    const int MT = NROI / 32;               // 8 M-strips of 32 rows

    // offset MLP: 12544 -> 1024 -> 1024 -> 98
    gemm_wmma_f32_k<<<dim3(1024 / 64, MT), blk, 0, stream>>>(p,  ow1, ob1, h1,   1024, KIN,  1);
    gemm_wmma_f32_k<<<dim3(1024 / 64, MT), blk, 0, stream>>>(h1, ow2, ob2, h2,   1024, 1024, 1);
    gemm_wmma_f32_k<<<dim3(128 / 64,  MT), blk, 0, stream>>>(h2, wp3, ob3, offb, 98,   1024, 0);

    // pass 2: deformable pool with learned offsets -> p (reuse buffer)
    droi_pool_k<<<NROI * NBIN, CCH, 0, stream>>>(feat, ro, offb, 1, p);

    // box head: 12544 -> 512 -> 512 -> 4
    gemm_wmma_f32_k<<<dim3(512 / 64, MT), blk, 0, stream>>>(p,  fw1, fb1, f1,  512, KIN, 1);
    gemm_wmma_f32_k<<<dim3(512 / 64, MT), blk, 0, stream>>>(f1, fw2, fb2, f2,  512, 512, 1);
    gemm_wmma_f32_k<<<dim3(64 / 64,  MT), blk, 0, stream>>>(f2, wpb, bxb, out, 4,   512, 0);
}